// YoloLayer_81020263072174
// MI455X (gfx1250) — compile-verified
//
#include <hip/hip_runtime.h>

// YOLO layer (B=64, nA=3, nC=3, G=152): pure memory-bound permute + elementwise.
// ~355 MB traffic -> ~15.2 us roofline @ 23.3 TB/s. No matmul -> no WMMA.
// CDNA5 paths used: global_load_async_to_lds_b128 staging (ASYNCcnt) + LDS
// transpose for coalesced AoS output + hw trans ops (v_exp_f32/v_rcp_f32).

#define G_   152
#define GG_  (G_ * G_)          // 23104
#define NA_  3
#define NB_  64
#define CH_  10                 // 7 + nC
#define TPB_ 256
#define PIXBLK_ ((GG_ + TPB_ - 1) / TPB_)   // 91

typedef int v4i __attribute__((ext_vector_type(4)));
typedef __attribute__((address_space(1))) v4i gv4i;   // global int4
typedef __attribute__((address_space(3))) v4i lv4i;   // LDS int4

__device__ __forceinline__ float fast_exp(float x) {
    return __builtin_amdgcn_exp2f(x * 1.44269504088896341f);
}
__device__ __forceinline__ float fast_sigmoid(float x) {
    return __builtin_amdgcn_rcpf(1.0f + __builtin_amdgcn_exp2f(-1.44269504088896341f * x));
}

__global__ __launch_bounds__(TPB_) void yolo_head_kernel(
    const float* __restrict__ x,
    const float* __restrict__ anchors,
    const int*   __restrict__ img_size,
    float* __restrict__ out)
{
    __shared__ __align__(16) float smem[TPB_ * CH_];   // 10 KB, reused in/out

    const int t    = threadIdx.x;
    const int by   = blockIdx.y;            // b*NA + a
    const int a    = by % NA_;
    const int p0   = blockIdx.x * TPB_;
    const int npix = min(TPB_, GG_ - p0);

    // ---- stage CH_ channel slabs (TPB_ floats each, contiguous) into LDS ----
    const float* chan_base = x + (size_t)(by * CH_) * GG_ + (size_t)p0;

#if __has_builtin(__builtin_amdgcn_global_load_async_to_lds_b128)
    for (int f = t * 4; f < TPB_ * CH_; f += TPB_ * 4) {
        const int c  = f / TPB_;
        const int o  = f - c * TPB_;
        const int oc = min(o, GG_ - 4 - p0);        // fully-OOB rows clamp in-range
        const float* gsrc = chan_base + (size_t)c * GG_ + oc;
        gv4i* gp = (gv4i*)(unsigned long long)gsrc;
        lv4i* lp = (lv4i*)(unsigned int)(unsigned long long)(&smem[f]);
        __builtin_amdgcn_global_load_async_to_lds_b128(gp, lp, 0, 0);
    }
  #if __has_builtin(__builtin_amdgcn_s_wait_asynccnt)
    __builtin_amdgcn_s_wait_asynccnt(0);
  #else
    asm volatile("s_wait_asynccnt 0" ::: "memory");
  #endif
#else
    for (int f = t * 4; f < TPB_ * CH_; f += TPB_ * 4) {
        const int c  = f / TPB_;
        const int o  = f - c * TPB_;
        const int oc = min(o, GG_ - 4 - p0);
        const float4 v = *(const float4*)(chan_base + (size_t)c * GG_ + oc);
        *(float4*)(&smem[f]) = v;
    }
#endif
    __syncthreads();

    // ---- compute: gather my pixel's 10 channels (bank-conflict-free) ----
    float v[CH_];
    const int  p     = p0 + t;
    const bool valid = (p < GG_);
    if (valid) {
        #pragma unroll
        for (int c = 0; c < CH_; ++c) v[c] = smem[c * TPB_ + t];
    }
    __syncthreads();   // before reusing smem as output staging

    if (valid) {
        const float stride = (float)img_size[0] / (float)G_;   // 608/152 = 4
        const float aw  = anchors[a * 4 + 0] / stride;
        const float ah  = anchors[a * 4 + 1] / stride;
        const float fgx = (float)(p % G_);
        const float fgy = (float)(p / G_);

        float* dst = &smem[t * CH_];            // stride-10 lanes: conflict-free
        dst[0] = (fast_sigmoid(v[0]) + fgx) * stride;
        dst[1] = (fast_sigmoid(v[1]) + fgy) * stride;
        dst[2] = fminf(fast_exp(v[2]), 1000.0f) * aw * stride;
        dst[3] = fminf(fast_exp(v[3]), 1000.0f) * ah * stride;
        dst[4] = v[4];
        dst[5] = v[5];
        dst[6] = fast_sigmoid(v[6]);
        dst[7] = fast_sigmoid(v[7]);
        dst[8] = fast_sigmoid(v[8]);
        dst[9] = fast_sigmoid(v[9]);
    }
    __syncthreads();

    // ---- coalesced AoS store: 512B-per-wave b128 bursts ----
    float* obase = out + ((size_t)by * GG_ + (size_t)p0) * CH_;
    const int nfloat = npix * CH_;
    const int n4     = nfloat >> 2;
    for (int i4 = t; i4 < n4; i4 += TPB_)
        ((float4*)obase)[i4] = ((const float4*)smem)[i4];
    for (int i = (n4 << 2) + t; i < nfloat; i += TPB_)
        obase[i] = smem[i];
}

extern "C" void kernel_launch(void* const* d_in, const int* in_sizes, int n_in,
                              void* d_out, int out_size, void* d_ws, size_t ws_size,
                              hipStream_t stream) {
    (void)in_sizes; (void)n_in; (void)out_size; (void)d_ws; (void)ws_size;
    const float* x       = (const float*)d_in[0];
    // d_in[1] = targets: 0 elements, unused by the reference output
    const float* anchors = (const float*)d_in[2];
    const int*   imgsz   = (const int*)d_in[3];
    float*       out     = (float*)d_out;

    dim3 grid(PIXBLK_, NB_ * NA_, 1);
    yolo_head_kernel<<<grid, TPB_, 0, stream>>>(x, anchors, imgsz, out);
}